// policy_5755256177435
// MI455X (gfx1250) — compile-verified
//
#include <hip/hip_runtime.h>

typedef __attribute__((ext_vector_type(2))) float v2f;
typedef __attribute__((ext_vector_type(8))) float v8f;

#define H 8192
#define KSPLIT 16

// d_ws layout (floats):
//   [0, H)        h1   = relu(state @ W1 + b1)
//   [H, 2H)       acc2 = b2 + h1 @ W2   (pre-relu accumulator)
//   [2H, 2H+8)    pen  = b3 + relu(acc2) @ W3

// ---------------- Layer 1 (42 -> 8192) + accumulator init ----------------
__global__ __launch_bounds__(256) void k_layer1(const float* __restrict__ state,
                                                const float* __restrict__ W1,
                                                const float* __restrict__ b1,
                                                const float* __restrict__ b2,
                                                const float* __restrict__ b3,
                                                float* __restrict__ ws) {
  int j = blockIdx.x * blockDim.x + threadIdx.x;   // 0..H-1
  float s = b1[j];
  #pragma unroll
  for (int k = 0; k < 42; ++k)
    s = fmaf(state[k], W1[(size_t)k * H + j], s);
  ws[j]     = fmaxf(s, 0.0f);   // h1 (post-relu)
  ws[H + j] = b2[j];            // acc2 init with bias
  if (j < 8) ws[2 * H + j] = b3[j];
}

// ---------------- Layer 2 (8192 -> 8192) via V_WMMA_F32_16X16X4_F32 ------
// Single-wave workgroup = one 16-column output tile x one 512-long K slice.
// A fragment: the SAME h1[k..k+3] chunk is broadcast into all 16 rows
// (every lane loads h1[k+2*half .. +1] as one b64 -> no lane predication).
// All rows of D then hold the identical, correct dot product; row 0
// (VGPR c[0], lanes 0..15) is used. B fragment: VGPR0 = W2 rows k / k+2,
// VGPR1 = rows k+1 / k+3, column n0 + lane%16 (coalesced 2x64B per load).
__global__ __launch_bounds__(32) void k_layer2_wmma(const float* __restrict__ W2,
                                                    float* __restrict__ ws) {
  const float* h1  = ws;
  float*      acc2 = ws + H;

  const int lane = threadIdx.x;        // 0..31
  const int n0   = blockIdx.x * 16;    // output column tile
  const int kPer = H / KSPLIT;         // 512
  const int kBeg = blockIdx.y * kPer;

  const int half = lane >> 4;          // 0: K=0,1   1: K=2,3
  const int nl   = lane & 15;

  v8f c = {};

  const float* wp = W2 + (size_t)(kBeg + half * 2) * H + n0 + nl;
  const float* ap = h1 + kBeg + half * 2;   // 8-byte aligned

  #pragma unroll 4
  for (int it = 0; it < kPer / 4; ++it) {
    v2f a = *(const v2f*)ap;           // h1[k+2h], h1[k+2h+1]  (L0-hot b64)
    v2f b;
    b.x = wp[0];                       // rows k / k+2
    b.y = wp[H];                       // rows k+1 / k+3
    // (neg_a, A, neg_b, B, c_mod, C, reuse_a, reuse_b)
    c = __builtin_amdgcn_wmma_f32_16x16x4_f32(false, a, false, b,
                                              (short)0, c, false, false);
    wp += 4 * H;
    ap += 4;
  }

  // D row 0 lives in VGPR c[0], lanes 0..15 (N = lane).
  if (lane < 16) atomicAdd(&acc2[n0 + lane], c[0]);
}

// ---------------- Layer 3 (8192 -> 8): relu + GEMV with wave reduction ---
__global__ __launch_bounds__(256) void k_layer3(const float* __restrict__ W3,
                                                float* __restrict__ ws) {
  const float* acc2 = ws + H;
  float*       pen  = ws + 2 * H;
  int j = blockIdx.x * blockDim.x + threadIdx.x;   // 0..H-1
  float hj = fmaxf(acc2[j], 0.0f);

  #pragma unroll
  for (int n = 0; n < 8; ++n) {
    float v = hj * W3[(size_t)j * 8 + n];
    #pragma unroll
    for (int off = 16; off > 0; off >>= 1)
      v += __shfl_xor(v, off, 32);                 // wave32 reduce
    if ((threadIdx.x & 31) == 0) atomicAdd(&pen[n], v);
  }
}

// ---------------- Head: value + masked softmax ---------------------------
__global__ __launch_bounds__(32) void k_head(const float* __restrict__ state,
                                             const float* __restrict__ ws,
                                             float* __restrict__ out) {
  if (threadIdx.x != 0) return;
  const float* pen = ws + 2 * H;
  float value = pen[7];

  float logits[7];
  bool  legal[7];
  float m = -1e30f;
  for (int i = 0; i < 7; ++i) {
    legal[i]  = (state[i] == 0.0f);   // state[0] is forced 0 -> always legal
    logits[i] = pen[i];
    if (legal[i] && logits[i] > m) m = logits[i];
  }
  float e[7], sum = 0.f;
  for (int i = 0; i < 7; ++i) {
    e[i] = legal[i] ? expf(logits[i] - m) : 0.0f;
    sum += e[i];
  }
  out[0] = value;
  for (int i = 0; i < 7; ++i) out[1 + i] = e[i] / sum;
}

extern "C" void kernel_launch(void* const* d_in, const int* in_sizes, int n_in,
                              void* d_out, int out_size, void* d_ws, size_t ws_size,
                              hipStream_t stream) {
  const float* state = (const float*)d_in[0];
  const float* W1    = (const float*)d_in[1];
  const float* b1    = (const float*)d_in[2];
  const float* W2    = (const float*)d_in[3];
  const float* b2    = (const float*)d_in[4];
  const float* W3    = (const float*)d_in[5];
  const float* b3    = (const float*)d_in[6];
  float* out = (float*)d_out;
  float* ws  = (float*)d_ws;

  k_layer1<<<H / 256, 256, 0, stream>>>(state, W1, b1, b2, b3, ws);
  k_layer2_wmma<<<dim3(H / 16, KSPLIT), 32, 0, stream>>>(W2, ws);
  k_layer3<<<H / 256, 256, 0, stream>>>(W3, ws);
  k_head<<<1, 32, 0, stream>>>(state, ws, out);
}